// GATModel_28965259444713
// MI455X (gfx1250) — compile-verified
//
#include <hip/hip_runtime.h>
#include <cstdint>

#define DEV __device__ __forceinline__

namespace {

constexpr int NN    = 20000;   // nodes
constexpr int NE    = 160000;  // edges (without self loops)
constexpr int ET    = 180000;  // edges + self loops
constexpr int FEAT  = 128;
constexpr int HID   = 64;
constexpr int HEADS = 8;
constexpr int HD    = 512;     // HEADS*HID
constexpr int G     = 64;      // graphs

typedef __attribute__((ext_vector_type(2))) float v2f;
typedef __attribute__((ext_vector_type(8))) float v8f;

DEV void atomicMaxF(float* addr, float v) {
  if (v >= 0.0f) atomicMax((int*)addr, __float_as_int(v));
  else           atomicMin((unsigned int*)addr, (unsigned int)__float_as_int(v));
}

DEV float waveSum(float v) {
#pragma unroll
  for (int off = 16; off > 0; off >>= 1) v += __shfl_xor(v, off, 32);
  return v;
}

DEV void edgeSD(const int* __restrict__ ei, int e, int& s, int& d) {
  if (e < NE) { s = ei[e]; d = ei[NE + e]; }
  else        { s = e - NE; d = e - NE; }
}

// ---------------- WMMA GEMM: C[M,N] = A[M,K] @ B[K,N], fp32, row-major -----
// Wave computes a 16x64 strip: one A fragment reused across 4 16x16 tiles.
// A frag (16x4):  lane(0-15)=row, VGPR v holds K = 2*hi + v   (ISA 7.12.2)
// B frag (4x16):  lane(0-15)=col, VGPR v holds K = 2*hi + v   (mirrored)
// C/D (16x16): VGPR i, lanes0-15 -> M=i, lanes16-31 -> M=8+i, N=lane&15
// All 9 loads of a k-step are issued before the first WMMA so the scheduler
// can stagger s_wait_loadcnt and overlap tile 1-3 load latency with tile 0.
__global__ __launch_bounds__(256) void k_gemm(const float* __restrict__ A,
                                              const float* __restrict__ B,
                                              float* __restrict__ C,
                                              int M, int K, int N) {
  const int lane = threadIdx.x & 31;
  const int hi   = lane >> 4;
  const int lr   = lane & 15;
  const int strips = N >> 6;
  const int tiles  = (M >> 4) * strips;
  const int gw = blockIdx.x * 8 + (threadIdx.x >> 5);
  if (gw >= tiles) return;
  const int tm = gw / strips;
  const int tn = (gw - tm * strips) << 6;
  const float* Arow = A + (size_t)((tm << 4) + lr) * K;
  v8f a0 = {}, a1 = {}, a2 = {}, a3 = {};
  for (int k0 = 0; k0 < K; k0 += 4) {
    const int ka = k0 + 2 * hi;
    const float* B0 = B + (size_t)ka * N + tn + lr;
    const float* B1 = B0 + N;
    v2f av; av.x = Arow[ka]; av.y = Arow[ka + 1];
    v2f b0, b1, b2, b3;
    b0.x = B0[0];  b0.y = B1[0];
    b1.x = B0[16]; b1.y = B1[16];
    b2.x = B0[32]; b2.y = B1[32];
    b3.x = B0[48]; b3.y = B1[48];
    a0 = __builtin_amdgcn_wmma_f32_16x16x4_f32(false, av, false, b0, (short)0, a0, false, false);
    a1 = __builtin_amdgcn_wmma_f32_16x16x4_f32(false, av, false, b1, (short)0, a1, false, false);
    a2 = __builtin_amdgcn_wmma_f32_16x16x4_f32(false, av, false, b2, (short)0, a2, false, false);
    a3 = __builtin_amdgcn_wmma_f32_16x16x4_f32(false, av, false, b3, (short)0, a3, false, false);
  }
  float* Cb = C + (size_t)((tm << 4) + 8 * hi) * N + tn + lr;
#pragma unroll
  for (int i = 0; i < 8; ++i) {
    float* r = Cb + (size_t)i * N;
    r[0] = a0[i]; r[16] = a1[i]; r[32] = a2[i]; r[48] = a3[i];
  }
}

// ---------------- elementwise / init ---------------------------------------
__global__ void k_fill(float* p, float v, int n) {
  int i = blockIdx.x * blockDim.x + threadIdx.x;
  if (i < n) p[i] = v;
}
__global__ void k_fill_rows(float* p, const float* __restrict__ bias, int n, int fmask) {
  int i = blockIdx.x * blockDim.x + threadIdx.x;
  if (i < n) p[i] = bias[i & fmask];
}
__global__ void k_count(const int* __restrict__ batch, float* ncnt) {
  int i = blockIdx.x * blockDim.x + threadIdx.x;
  if (i < NN) atomicAdd(&ncnt[batch[i]], 1.0f);
}

// ---------------- GAT attention --------------------------------------------
__global__ void k_attn(const float* __restrict__ h, const float* __restrict__ aS,
                       const float* __restrict__ aD, float* es, float* ed) {
  int i = blockIdx.x * blockDim.x + threadIdx.x;
  if (i >= NN * HEADS) return;
  int n = i >> 3, hd = i & 7;
  const float* hr = h + (size_t)n * HD + hd * HID;
  const float* as = aS + hd * HID;
  const float* ad = aD + hd * HID;
  float ss = 0.f, sd = 0.f;
  for (int c = 0; c < HID; ++c) { float v = hr[c]; ss += v * as[c]; sd += v * ad[c]; }
  es[i] = ss; ed[i] = sd;
}

DEV float edge_logit(const float* es, const float* ed, int s, int d, int hd) {
  float v = es[s * 8 + hd] + ed[d * 8 + hd];
  return v > 0.f ? v : 0.2f * v;            // leaky_relu 0.2
}

__global__ void k_edge_max(const int* __restrict__ ei, const float* __restrict__ es,
                           const float* __restrict__ ed, float* mx) {
  int i = blockIdx.x * blockDim.x + threadIdx.x;
  if (i >= ET * HEADS) return;
  int e = i >> 3, hd = i & 7, s, d;
  edgeSD(ei, e, s, d);
  atomicMaxF(&mx[d * 8 + hd], edge_logit(es, ed, s, d, hd));
}

__global__ void k_edge_expsum(const int* __restrict__ ei, const float* __restrict__ es,
                              const float* __restrict__ ed, const float* __restrict__ mx,
                              float* sx, float* wE) {
  int i = blockIdx.x * blockDim.x + threadIdx.x;
  if (i >= ET * HEADS) return;
  int e = i >> 3, hd = i & 7, s, d;
  edgeSD(ei, e, s, d);
  float ex = expf(edge_logit(es, ed, s, d, hd) - mx[d * 8 + hd]);
  wE[i] = ex;
  atomicAdd(&sx[d * 8 + hd], ex);
}

__global__ void k_edge_agg(const int* __restrict__ ei, const float* __restrict__ wE,
                           const float* __restrict__ sx, const float* __restrict__ hm,
                           float* out) {
  int i = blockIdx.x * blockDim.x + threadIdx.x;
  if (i >= ET * 128) return;                 // 128 float4-groups of the 512-wide row
  int e = i >> 7;
  int q = (i & 127) << 2;
  int hd = q >> 6;
  int s, d;
  edgeSD(ei, e, s, d);
  float alpha = wE[e * 8 + hd] / (sx[d * 8 + hd] + 1e-16f);
  const float4 hv = *(const float4*)(hm + (size_t)s * HD + q);
  float* o = out + (size_t)d * HD + q;
  atomicAdd(o + 0, alpha * hv.x);
  atomicAdd(o + 1, alpha * hv.y);
  atomicAdd(o + 2, alpha * hv.z);
  atomicAdd(o + 3, alpha * hv.w);
}

__global__ void k_head_mean(const float* __restrict__ agg, const float* __restrict__ b3,
                            float* out) {
  int i = blockIdx.x * blockDim.x + threadIdx.x;
  if (i >= NN * HID) return;
  int n = i >> 6, c = i & 63;
  const float* r = agg + (size_t)n * HD + c;
  float s = 0.f;
#pragma unroll
  for (int hd = 0; hd < HEADS; ++hd) s += r[hd * HID];
  out[i] = s * 0.125f + b3[c];
}

// ---------------- graph LayerNorm (mode='graph') ---------------------------
__global__ __launch_bounds__(256) void k_ln_sum(const float* __restrict__ x,
                                                const int* __restrict__ batch,
                                                float* gsum, int F) {
  int wv = (blockIdx.x * 256 + threadIdx.x) >> 5;
  if (wv >= NN) return;
  int lane = threadIdx.x & 31;
  const float* xr = x + (size_t)wv * F;
  float s = 0.f;
  for (int i = lane; i < F; i += 32) s += xr[i];
  s = waveSum(s);
  if (lane == 0) atomicAdd(&gsum[batch[wv]], s);
}

__global__ __launch_bounds__(256) void k_ln_var(const float* __restrict__ x,
                                                const int* __restrict__ batch,
                                                const float* __restrict__ gsum,
                                                const float* __restrict__ ncnt,
                                                float* gvar, int F) {
  int wv = (blockIdx.x * 256 + threadIdx.x) >> 5;
  if (wv >= NN) return;
  int lane = threadIdx.x & 31;
  int g = batch[wv];
  float mean = gsum[g] / (ncnt[g] * (float)F);
  const float* xr = x + (size_t)wv * F;
  float s = 0.f;
  for (int i = lane; i < F; i += 32) { float dd = xr[i] - mean; s += dd * dd; }
  s = waveSum(s);
  if (lane == 0) atomicAdd(&gvar[g], s);
}

__global__ void k_ln_apply(const float* __restrict__ x, const int* __restrict__ batch,
                           const float* __restrict__ gsum, const float* __restrict__ gvar,
                           const float* __restrict__ ncnt, const float* __restrict__ w,
                           const float* __restrict__ b, float* y,
                           int fshift, int fmask, int ntot) {
  int i = blockIdx.x * blockDim.x + threadIdx.x;
  if (i >= ntot) return;
  int n = i >> fshift, f = i & fmask;
  int g = batch[n];
  float cf = ncnt[g] * (float)(fmask + 1);
  float mean = gsum[g] / cf;
  float var  = gvar[g] / cf;
  float rs = rsqrtf(var + 1e-5f);
  float v = (x[i] - mean) * rs * w[f] + b[f];
  y[i] = fmaxf(v, 0.f);                     // fused ReLU (ref: relu(gln(...)))
}

// ---------------- GCN branch -----------------------------------------------
__global__ void k_deg(const int* __restrict__ ei, float* deg) {
  int e = blockIdx.x * blockDim.x + threadIdx.x;
  if (e >= ET) return;
  int s, d;
  edgeSD(ei, e, s, d);
  atomicAdd(&deg[d], 1.0f);
}

__global__ void k_gcn_agg(const int* __restrict__ ei, const float* __restrict__ deg,
                          const float* __restrict__ hg, float* outg) {
  int i = blockIdx.x * blockDim.x + threadIdx.x;
  if (i >= ET * 16) return;                  // 16 float4-groups of the 64-wide row
  int e = i >> 4;
  int q = (i & 15) << 2;
  int s, d;
  edgeSD(ei, e, s, d);
  float norm = rsqrtf(deg[s]) * rsqrtf(deg[d]);
  const float4 hv = *(const float4*)(hg + (size_t)s * HID + q);
  float* o = outg + (size_t)d * HID + q;
  atomicAdd(o + 0, norm * hv.x);
  atomicAdd(o + 1, norm * hv.y);
  atomicAdd(o + 2, norm * hv.z);
  atomicAdd(o + 3, norm * hv.w);
}

__global__ void k_relu_segmax(const float* __restrict__ x, const int* __restrict__ batch,
                              float* z) {
  int i = blockIdx.x * blockDim.x + threadIdx.x;
  if (i >= NN * HID) return;
  int n = i >> 6, c = i & 63;
  atomicMaxF(&z[batch[n] * HID + c], fmaxf(x[i], 0.f));
}

// ---------------- readout: z copy + tiny MLP -------------------------------
__global__ void k_copy_z(const float* __restrict__ z1, const float* __restrict__ z2,
                         float* out) {
  int i = blockIdx.x * blockDim.x + threadIdx.x;
  if (i >= G * 128) return;
  int g = i >> 7, c = i & 127;
  out[128 + i] = (c < 64) ? z1[g * 64 + c] : z2[g * 64 + (c - 64)];
}

__global__ __launch_bounds__(256) void k_mlp(const float* __restrict__ z1,
                                             const float* __restrict__ z2,
                                             const float* __restrict__ Wl0, const float* __restrict__ bl0,
                                             const float* __restrict__ Wl1, const float* __restrict__ bl1,
                                             const float* __restrict__ Wl2, const float* __restrict__ bl2,
                                             const float* __restrict__ Wo,  const float* __restrict__ bo,
                                             float* out) {
  __shared__ float ta[G * HID];
  __shared__ float tb[G * HID];
  int tid = threadIdx.x;
  for (int idx = tid; idx < G * HID; idx += 256) {
    int g = idx >> 6, j = idx & 63;
    float acc = bl0[j];
    const float* za = z1 + g * 64;
    const float* zb = z2 + g * 64;
    for (int k = 0; k < 64; ++k) acc += za[k] * Wl0[k * 64 + j];
    for (int k = 0; k < 64; ++k) acc += zb[k] * Wl0[(64 + k) * 64 + j];
    ta[idx] = fmaxf(acc, 0.f);
  }
  __syncthreads();
  for (int idx = tid; idx < G * HID; idx += 256) {
    int g = idx >> 6, j = idx & 63;
    float acc = bl1[j];
    for (int k = 0; k < 64; ++k) acc += ta[g * 64 + k] * Wl1[k * 64 + j];
    tb[idx] = fmaxf(acc, 0.f);
  }
  __syncthreads();
  for (int idx = tid; idx < G * HID; idx += 256) {
    int g = idx >> 6, j = idx & 63;
    float acc = bl2[j];
    for (int k = 0; k < 64; ++k) acc += tb[g * 64 + k] * Wl2[k * 64 + j];
    ta[idx] = fmaxf(acc, 0.f);
  }
  __syncthreads();
  for (int idx = tid; idx < G * 2; idx += 256) {
    int g = idx >> 1, o = idx & 1;
    float acc = bo[o];
    for (int k = 0; k < 64; ++k) acc += ta[g * 64 + k] * Wo[k * 2 + o];
    out[g * 2 + o] = acc;
  }
}

} // namespace

extern "C" void kernel_launch(void* const* d_in, const int* in_sizes, int n_in,
                              void* d_out, int out_size, void* d_ws, size_t ws_size,
                              hipStream_t stream) {
  (void)in_sizes; (void)n_in; (void)out_size; (void)ws_size;

  const float* x   = (const float*)d_in[0];
  const int*   ei  = (const int*)d_in[1];
  const int*  batch= (const int*)d_in[2];
  const float* W1  = (const float*)d_in[3];
  const float* aS1 = (const float*)d_in[4];
  const float* aD1 = (const float*)d_in[5];
  const float* b1  = (const float*)d_in[6];
  const float* W2  = (const float*)d_in[7];
  const float* aS2 = (const float*)d_in[8];
  const float* aD2 = (const float*)d_in[9];
  const float* b2  = (const float*)d_in[10];
  const float* W3  = (const float*)d_in[11];
  const float* aS3 = (const float*)d_in[12];
  const float* aD3 = (const float*)d_in[13];
  const float* b3  = (const float*)d_in[14];
  const float* Wg  = (const float*)d_in[15];
  const float* bg  = (const float*)d_in[16];
  const float* nw1 = (const float*)d_in[17];
  const float* nb1 = (const float*)d_in[18];
  const float* nw2 = (const float*)d_in[19];
  const float* nb2 = (const float*)d_in[20];
  const float* nw3 = (const float*)d_in[21];
  const float* nb3 = (const float*)d_in[22];
  const float* Wl0 = (const float*)d_in[23];
  const float* bl0 = (const float*)d_in[24];
  const float* Wl1 = (const float*)d_in[25];
  const float* bl1 = (const float*)d_in[26];
  const float* Wl2 = (const float*)d_in[27];
  const float* bl2 = (const float*)d_in[28];
  const float* Wo  = (const float*)d_in[29];
  const float* bo  = (const float*)d_in[30];
  float* out = (float*)d_out;

  // ---- workspace carve (floats) -------------------------------------------
  float* ws   = (float*)d_ws;
  float* big0 = ws;  ws += (size_t)NN * HD;     // h / activations ping
  float* big1 = ws;  ws += (size_t)NN * HD;     // h / activations pong
  float* sm0  = ws;  ws += (size_t)NN * HID;    // layer-3 mean+bias
  float* sm1  = ws;  ws += (size_t)NN * HID;    // layer-3 LN output
  float* hg   = ws;  ws += (size_t)NN * HID;    // GCN x@Wg
  float* og   = ws;  ws += (size_t)NN * HID;    // GCN aggregated
  float* es   = ws;  ws += (size_t)NN * HEADS;
  float* ed   = ws;  ws += (size_t)NN * HEADS;
  float* mx   = ws;  ws += (size_t)NN * HEADS;
  float* sx   = ws;  ws += (size_t)NN * HEADS;
  float* wE   = ws;  ws += (size_t)ET * HEADS;  // exp weights per edge/head
  float* deg  = ws;  ws += NN;
  float* gsum = ws;  ws += G;
  float* gvar = ws;  ws += G;
  float* ncnt = ws;  ws += G;
  float* z1   = ws;  ws += (size_t)G * HID;
  float* z2   = ws;  ws += (size_t)G * HID;

  const float NEG = -3.402823466e+38f;
  auto cd = [](long long a, int b) { return (int)((a + b - 1) / b); };

  auto gat_attention = [&](const float* hmat, const float* aS, const float* aD,
                           float* agg_out) {
    k_attn<<<cd((long long)NN * HEADS, 256), 256, 0, stream>>>(hmat, aS, aD, es, ed);
    k_fill<<<cd((long long)NN * HEADS, 256), 256, 0, stream>>>(mx, NEG, NN * HEADS);
    k_fill<<<cd((long long)NN * HEADS, 256), 256, 0, stream>>>(sx, 0.f, NN * HEADS);
    k_edge_max<<<cd((long long)ET * HEADS, 256), 256, 0, stream>>>(ei, es, ed, mx);
    k_edge_expsum<<<cd((long long)ET * HEADS, 256), 256, 0, stream>>>(ei, es, ed, mx, sx, wE);
    k_edge_agg<<<cd((long long)ET * 128, 256), 256, 0, stream>>>(ei, wE, sx, hmat, agg_out);
  };

  auto graph_ln = [&](const float* xin, const float* w, const float* b, float* yout,
                      int F, int fshift) {
    k_fill<<<1, G, 0, stream>>>(gsum, 0.f, G);
    k_fill<<<1, G, 0, stream>>>(gvar, 0.f, G);
    k_ln_sum<<<cd((long long)NN * 32, 256), 256, 0, stream>>>(xin, batch, gsum, F);
    k_ln_var<<<cd((long long)NN * 32, 256), 256, 0, stream>>>(xin, batch, gsum, ncnt, gvar, F);
    k_ln_apply<<<cd((long long)NN * F, 256), 256, 0, stream>>>(xin, batch, gsum, gvar, ncnt,
                                                               w, b, yout, fshift, F - 1, NN * F);
  };

  // ---- node counts per graph (shared by all LayerNorms) -------------------
  k_fill<<<1, G, 0, stream>>>(ncnt, 0.f, G);
  k_count<<<cd(NN, 256), 256, 0, stream>>>(batch, ncnt);

  // ---- GCN branch ---------------------------------------------------------
  k_gemm<<<cd((NN / 16) * (HID / 64), 8), 256, 0, stream>>>(x, Wg, hg, NN, FEAT, HID);
  k_fill<<<cd(NN, 256), 256, 0, stream>>>(deg, 0.f, NN);
  k_deg<<<cd(ET, 256), 256, 0, stream>>>(ei, deg);
  k_fill_rows<<<cd((long long)NN * HID, 256), 256, 0, stream>>>(og, bg, NN * HID, HID - 1);
  k_gcn_agg<<<cd((long long)ET * 16, 256), 256, 0, stream>>>(ei, deg, hg, og);
  k_fill<<<cd(G * HID, 256), 256, 0, stream>>>(z2, NEG, G * HID);
  k_relu_segmax<<<cd((long long)NN * HID, 256), 256, 0, stream>>>(og, batch, z2);

  // ---- GAT layer 1: x(128) -> 512 ----------------------------------------
  k_gemm<<<cd((NN / 16) * (HD / 64), 8), 256, 0, stream>>>(x, W1, big0, NN, FEAT, HD);
  k_fill_rows<<<cd((long long)NN * HD, 256), 256, 0, stream>>>(big1, b1, NN * HD, HD - 1);
  gat_attention(big0, aS1, aD1, big1);
  graph_ln(big1, nw1, nb1, big0, HD, 9);

  // ---- GAT layer 2: 512 -> 512 -------------------------------------------
  k_gemm<<<cd((NN / 16) * (HD / 64), 8), 256, 0, stream>>>(big0, W2, big1, NN, HD, HD);
  k_fill_rows<<<cd((long long)NN * HD, 256), 256, 0, stream>>>(big0, b2, NN * HD, HD - 1);
  gat_attention(big1, aS2, aD2, big0);
  graph_ln(big0, nw2, nb2, big1, HD, 9);

  // ---- GAT layer 3: 512 -> 512, mean over heads -> 64 ---------------------
  k_gemm<<<cd((NN / 16) * (HD / 64), 8), 256, 0, stream>>>(big1, W3, big0, NN, HD, HD);
  k_fill<<<cd((long long)NN * HD, 256), 256, 0, stream>>>(big1, 0.f, NN * HD);
  gat_attention(big0, aS3, aD3, big1);
  k_head_mean<<<cd((long long)NN * HID, 256), 256, 0, stream>>>(big1, b3, sm0);
  graph_ln(sm0, nw3, nb3, sm1, HID, 6);
  k_fill<<<cd(G * HID, 256), 256, 0, stream>>>(z1, NEG, G * HID);
  k_relu_segmax<<<cd((long long)NN * HID, 256), 256, 0, stream>>>(sm1, batch, z1);

  // ---- readout ------------------------------------------------------------
  k_copy_z<<<cd(G * 128, 256), 256, 0, stream>>>(z1, z2, out);
  k_mlp<<<1, 256, 0, stream>>>(z1, z2, Wl0, bl0, Wl1, bl1, Wl2, bl2, Wo, bo, out);
}